// PowerSpectrumFeatures_16226386444656
// MI455X (gfx1250) — compile-verified
//
#include <hip/hip_runtime.h>

// CDNA5 (gfx1250) power-spectrum features:
//   per atom, per l: G_l = cg_l * C_l^T C_l   with C_l : (2l+1) x 32  (f32)
//   output element (a,b,q,e,l) = G_l[a*8+q, b*8+e], l innermost.
// One wave32 per atom. Gram via V_WMMA_F32_16X16X4_F32 (exact f32 math).
// Bandwidth-bound: ~92MB in, ~492MB out -> non-temporal streaming both ways;
// LDS-staged l-interleave so the dominant output traffic is coalesced b128.

typedef float v2f __attribute__((ext_vector_type(2)));
typedef float v4f __attribute__((ext_vector_type(4)));
typedef float v8f __attribute__((ext_vector_type(8)));

#define WAVES_PER_BLOCK 4
#define ATOM_FLOATS 6144  // 4*4*8*8*6 output floats per atom (24 KB)

// One l-channel: accumulate G = C^T C with 16x16x4 f32 WMMA, scale by cg,
// scatter into LDS at l-interleaved offsets. K = 2L+1 is compile-time.
template <int L>
__device__ __forceinline__ void gram_channel(const float* __restrict__ C,
                                             float* __restrict__ my,
                                             int col, int khalf) {
    constexpr int K = 2 * L + 1;
    constexpr float SQ[6] = {1.0f, 0.57735026919f, 0.44721359550f,
                             0.37796447301f, 0.33333333333f, 0.30151134457f};
    constexpr float CG = (L & 1) ? -SQ[L] : SQ[L];

    v8f acc00 = {}, acc01 = {}, acc10 = {}, acc11 = {};

#pragma unroll
    for (int k0 = 0; k0 < K; k0 += 4) {
        const int k = k0 + 2 * khalf;  // this lane-half's first K index
        // Gram symmetry: one fragment serves as both A (row-block) and
        // B (col-block).  16x4 f32 A layout: lanes<16 hold K=k0,k0+1;
        // lanes>=16 hold K=k0+2,k0+3; zero-pad K >= 2L+1.
        v2f flo = {0.0f, 0.0f};  // columns 0..15 of C
        v2f fhi = {0.0f, 0.0f};  // columns 16..31 of C
        if (k < K) {
            flo.x = __builtin_nontemporal_load(C + k * 32 + col);
            fhi.x = __builtin_nontemporal_load(C + k * 32 + 16 + col);
        }
        if (k + 1 < K) {
            flo.y = __builtin_nontemporal_load(C + (k + 1) * 32 + col);
            fhi.y = __builtin_nontemporal_load(C + (k + 1) * 32 + 16 + col);
        }
        acc00 = __builtin_amdgcn_wmma_f32_16x16x4_f32(
            false, flo, false, flo, (short)0, acc00, false, false);
        acc01 = __builtin_amdgcn_wmma_f32_16x16x4_f32(
            false, flo, false, fhi, (short)0, acc01, false, false);
        acc10 = __builtin_amdgcn_wmma_f32_16x16x4_f32(
            false, fhi, false, flo, (short)0, acc10, false, false);
        acc11 = __builtin_amdgcn_wmma_f32_16x16x4_f32(
            false, fhi, false, fhi, (short)0, acc11, false, false);
    }

    // Scale by cg_l, scatter into LDS at l-interleaved offsets.
    // D layout: VGPR v, lane: M = v + 8*khalf, N = col.
#pragma unroll
    for (int t = 0; t < 4; ++t) {
        const int rb = (t >> 1) * 16;
        const int cb = (t & 1) * 16;
        const v8f acc = (t == 0) ? acc00 : (t == 1) ? acc01
                      : (t == 2) ? acc10 : acc11;
        const int c = cb + col;  // 0..31 = b*8+e
        const int b = c >> 3, e = c & 7;
#pragma unroll
        for (int v = 0; v < 8; ++v) {
            const int r = rb + v + 8 * khalf;  // 0..31 = a*8+q
            const int a = r >> 3, q = r & 7;
            const int idx = ((((a * 4 + b) * 8 + q) * 8 + e) * 6) + L;
            my[idx] = CG * acc[v];
        }
    }
}

__global__ __launch_bounds__(32 * WAVES_PER_BLOCK)
void ps_features_wmma(const float* __restrict__ s0, const float* __restrict__ s1,
                      const float* __restrict__ s2, const float* __restrict__ s3,
                      const float* __restrict__ s4, const float* __restrict__ s5,
                      float* __restrict__ out, int nAtoms) {
    __shared__ float lds[WAVES_PER_BLOCK * ATOM_FLOATS];

    const int lane  = threadIdx.x & 31;
    const int wave  = threadIdx.x >> 5;
    const int atom  = blockIdx.x * WAVES_PER_BLOCK + wave;
    const int col   = lane & 15;  // fragment column within 16-block
    const int khalf = lane >> 4;  // which K-pair this lane-half holds

    float* my = lds + wave * ATOM_FLOATS;

    if (atom < nAtoms) {  // wave-uniform: EXEC all-1s around WMMA
        gram_channel<0>(s0 + (size_t)atom * (1  * 32), my, col, khalf);
        gram_channel<1>(s1 + (size_t)atom * (3  * 32), my, col, khalf);
        gram_channel<2>(s2 + (size_t)atom * (5  * 32), my, col, khalf);
        gram_channel<3>(s3 + (size_t)atom * (7  * 32), my, col, khalf);
        gram_channel<4>(s4 + (size_t)atom * (9  * 32), my, col, khalf);
        gram_channel<5>(s5 + (size_t)atom * (11 * 32), my, col, khalf);
    }

    __syncthreads();  // order cross-lane LDS scatter before coalesced drain

    if (atom < nAtoms) {
        // 6144 floats -> 48 b128 per lane-strided loop; non-temporal stores
        // (output is write-once, never re-read; don't thrash the 192MB L2).
        v4f* __restrict__ dst = (v4f*)(out + (size_t)atom * ATOM_FLOATS);
        const v4f* __restrict__ src = (const v4f*)my;
#pragma unroll 4
        for (int j = lane; j < ATOM_FLOATS / 4; j += 32) {
            __builtin_nontemporal_store(src[j], dst + j);
        }
    }
}

extern "C" void kernel_launch(void* const* d_in, const int* in_sizes, int n_in,
                              void* d_out, int out_size, void* d_ws, size_t ws_size,
                              hipStream_t stream) {
    (void)n_in; (void)out_size; (void)d_ws; (void)ws_size;
    const float* s0 = (const float*)d_in[0];
    const float* s1 = (const float*)d_in[1];
    const float* s2 = (const float*)d_in[2];
    const float* s3 = (const float*)d_in[3];
    const float* s4 = (const float*)d_in[4];
    const float* s5 = (const float*)d_in[5];
    float* out = (float*)d_out;

    const int nAtoms = in_sizes[0] / 32;  // spex_0: (N,1,4,8)
    const int blocks = (nAtoms + WAVES_PER_BLOCK - 1) / WAVES_PER_BLOCK;
    ps_features_wmma<<<blocks, 32 * WAVES_PER_BLOCK, 0, stream>>>(
        s0, s1, s2, s3, s4, s5, out, nAtoms);
}